// LocallyConnected2D_3710851744184
// MI455X (gfx1250) — compile-verified
//
#include <hip/hip_runtime.h>
#include <stdint.h>

typedef __attribute__((ext_vector_type(2))) float v2f;
typedef __attribute__((ext_vector_type(8))) float v8f;

#define B_   64
#define H_   32
#define W_   32
#define C_   64
#define O_   64
#define OR_  30
#define OC_  30
#define KW_  3
#define F_   576      // 3*3*C
#define KC_  64       // k-chunk = C_, one (di,dj) tap per chunk
#define NCHUNK_ 9
#define AS_STRIDE 68  // padded LDS row stride (bank-conflict-free, 16B aligned)

// ---- CDNA5 async global->LDS (ASYNCcnt-tracked, no VGPR round trip) ----
__device__ __forceinline__ void async_copy_b128(uint32_t lds_off, const float* g) {
  asm volatile("global_load_async_to_lds_b128 %0, %1, off"
               :: "v"(lds_off), "v"(g) : "memory");
}
__device__ __forceinline__ void wait_async0() {
  asm volatile("s_wait_asynccnt 0x0" ::: "memory");
}
__device__ __forceinline__ uint32_t lds_off(const float* p) {
  // generic LDS-aperture address: addr[31:0] == LDS byte offset
  return (uint32_t)(uintptr_t)p;
}

__device__ __forceinline__ void stage_chunk(const float* __restrict__ x,
                                            const float* __restrict__ kp,
                                            int q, int r, int cc, int tid,
                                            float* AsB, float* WsB) {
  const int di = q / KW_;
  const int dj = q % KW_;
  #pragma unroll
  for (int i = 0; i < 4; ++i) {
    const int idx = tid + i * 256;       // 0..1023
    const int row = idx >> 4;            // 0..63
    const int c4  = (idx & 15) << 2;     // 0,4,...,60
    // A: x[row, r+di, cc+dj, c4..c4+3]  (16B aligned, coalesced)
    const float* ga = x + ((((size_t)row * H_ + (r + di)) * W_ + (cc + dj)) * C_ + c4);
    async_copy_b128(lds_off(AsB + row * AS_STRIDE + c4), ga);
    // W: kern[p, q*64+row, c4..c4+3]    (contiguous 16KB stream)
    const float* gw = kp + ((size_t)(q * KC_ + row) * O_ + c4);
    async_copy_b128(lds_off(WsB + row * O_ + c4), gw);
  }
}

__global__ __launch_bounds__(256) void lc2d_wmma_f32(
    const float* __restrict__ x,      // (B,H,W,C)
    const float* __restrict__ kern,   // (P,F,O)
    const float* __restrict__ bias,   // (OR,OC,O)
    float* __restrict__ out)          // (B,OR,OC,O)
{
  __shared__ float As[2][B_ * AS_STRIDE];   // double-buffered patch slices
  __shared__ float Ws[2][KC_ * O_];         // double-buffered weight slices

  const int p    = blockIdx.x;           // 0..899
  const int r    = p / OC_;
  const int cc   = p % OC_;
  const int tid  = threadIdx.x;
  const int lane = tid & 31;
  const int wave = tid >> 5;             // 0..7
  const int tn   = wave & 3;             // filter tile: n0 = tn*16
  const int tmh  = wave >> 2;            // batch half: m tiles {tmh*32, tmh*32+16}
  const int n0   = tn * 16;

  const int l15  = lane & 15;
  const int hi   = lane >> 4;            // 0: lanes 0-15, 1: lanes 16-31
  const int koff = hi * 2;               // ISA 16x4 f32 A/B layout K offset
  const int m0   = tmh * 32 + l15;       // A-frag row, tile 0
  const int m1   = m0 + 16;              // A-frag row, tile 1

  const float* kp = kern + (size_t)p * F_ * O_;

  v8f acc0 = {};
  v8f acc1 = {};

  // prologue: start chunk 0 streaming while nothing else is in flight
  stage_chunk(x, kp, 0, r, cc, tid, As[0], Ws[0]);

  for (int q = 0; q < NCHUNK_; ++q) {
    wait_async0();        // my async writes for chunk q have landed in LDS
    __syncthreads();      // everyone's have

    if (q + 1 < NCHUNK_)  // prefetch next chunk into the other buffer
      stage_chunk(x, kp, q + 1, r, cc, tid, As[(q + 1) & 1], Ws[(q + 1) & 1]);

    const float* AsB = As[q & 1];
    const float* WsB = Ws[q & 1];

    #pragma unroll
    for (int kk = 0; kk < KC_; kk += 4) {
      const int ka = kk + koff;
      // A frags: contiguous K pair -> ds_load_b64 each
      v2f a0 = *(const v2f*)(&AsB[m0 * AS_STRIDE + ka]);
      v2f a1 = *(const v2f*)(&AsB[m1 * AS_STRIDE + ka]);
      // B frag: {W[ka][n], W[ka+1][n]} -> single ds_load_2addr_b32 offset1:64
      const float bx = WsB[(ka    ) * O_ + n0 + l15];
      const float by = WsB[(ka + 1) * O_ + n0 + l15];
      v2f b; b.x = bx; b.y = by;

      acc0 = __builtin_amdgcn_wmma_f32_16x16x4_f32(false, a0, false, b,
                                                   (short)0, acc0, false, false);
      acc1 = __builtin_amdgcn_wmma_f32_16x16x4_f32(false, a1, false, b,
                                                   (short)0, acc1, false, false);
    }
    __syncthreads();      // done reading buf q before it is restaged (q+2)
  }

  // ---- epilogue: bias + ReLU + store (C/D layout: VGPR j -> M=j / j+8) ----
  const int n  = n0 + l15;
  const float bv = bias[(r * OC_ + cc) * O_ + n];
  #pragma unroll
  for (int t = 0; t < 2; ++t) {
    const v8f acc = t ? acc1 : acc0;
    #pragma unroll
    for (int j = 0; j < 8; ++j) {
      const int m = tmh * 32 + t * 16 + j + hi * 8;   // global batch row
      out[(((size_t)m * OR_ + r) * OC_ + cc) * O_ + n] = fmaxf(acc[j] + bv, 0.0f);
    }
  }
}

extern "C" void kernel_launch(void* const* d_in, const int* in_sizes, int n_in,
                              void* d_out, int out_size, void* d_ws, size_t ws_size,
                              hipStream_t stream) {
  (void)in_sizes; (void)n_in; (void)out_size; (void)d_ws; (void)ws_size;
  const float* x    = (const float*)d_in[0];
  const float* kern = (const float*)d_in[1];
  const float* bias = (const float*)d_in[2];
  float* out        = (float*)d_out;

  dim3 grid(OR_ * OC_);   // 900 locations, one workgroup each
  dim3 block(256);        // 8 wave32s
  lc2d_wmma_f32<<<grid, block, 0, stream>>>(x, kern, bias, out);
}